// Main_model_15032385536152
// MI455X (gfx1250) — compile-verified
//
#include <hip/hip_runtime.h>
#include <math.h>

typedef __attribute__((ext_vector_type(2))) float v2f;
typedef __attribute__((ext_vector_type(8))) float v8f;

static __device__ __forceinline__ float sigmoidf_(float x) { return 1.f / (1.f + __expf(-x)); }

// ---------------- fp32 WMMA GEMM:  C[M,N] = A[M,K] @ B[K,N] (+bias)(+relu) ----------------
// One wave32 computes a 16x64 output tile: one shared A fragment drives 4 independent
// v_wmma_f32_16x16x4_f32 accumulator chains (ILP + A-load amortization).
// Requires N % 64 == 0 (all layer widths here are 64/128/256). Ragged M handled by
// index clamping (no predicated loads -> no EXEC save/restore in the hot loop).
__global__ void gemm_wmma_f32(const float* __restrict__ A,
                              const float* __restrict__ B,
                              const float* __restrict__ bias,
                              float* __restrict__ C,
                              int M, int K, int N, int relu)
{
    const int  lane    = threadIdx.x & 31;
    const long wave    = ((long)blockIdx.x * blockDim.x + threadIdx.x) >> 5;
    const int  groupsN = N >> 6;
    const long tilesM  = (M + 15) >> 4;
    if (wave >= tilesM * (long)groupsN) return;
    const int tm   = (int)(wave / groupsN);
    const int gn   = (int)(wave % groupsN);
    const int half = lane >> 4;          // 0: K rows {k,k+1} / 1: {k+2,k+3}
    const int l16  = lane & 15;
    const int row0 = tm << 4;
    const int col0 = gn << 6;

    int ar = row0 + l16;                 // clamp instead of predicating the loads
    if (ar >= M) ar = M - 1;
    const float* pA = A + (long)ar * K + (half << 1);
    const float* pB = B + (long)(half << 1) * N + col0 + l16;
    const long  strideB = 4L * (long)N;

    v8f acc0 = {}, acc1 = {}, acc2 = {}, acc3 = {};
#pragma unroll 2
    for (int k = 0; k < K; k += 4) {
        const v2f a = *(const v2f*)pA;                 // 8B A fragment
        v2f b0, b1, b2, b3;
        b0.x = pB[0];  b0.y = pB[N];                   // v0 rows {k,k+2}, v1 rows {k+1,k+3}
        b1.x = pB[16]; b1.y = pB[N + 16];
        b2.x = pB[32]; b2.y = pB[N + 32];
        b3.x = pB[48]; b3.y = pB[N + 48];
        acc0 = __builtin_amdgcn_wmma_f32_16x16x4_f32(false, a, false, b0, (short)0, acc0, false, false);
        acc1 = __builtin_amdgcn_wmma_f32_16x16x4_f32(false, a, false, b1, (short)0, acc1, false, false);
        acc2 = __builtin_amdgcn_wmma_f32_16x16x4_f32(false, a, false, b2, (short)0, acc2, false, false);
        acc3 = __builtin_amdgcn_wmma_f32_16x16x4_f32(false, a, false, b3, (short)0, acc3, false, false);
        pA += 4;
        pB += strideB;
    }
#pragma unroll
    for (int r = 0; r < 8; ++r) {
        const int row = row0 + r + (half << 3);
        if (row < M) {
            const long base = (long)row * N + col0 + l16;
            float v0 = acc0[r], v1 = acc1[r], v2 = acc2[r], v3 = acc3[r];
            if (bias) {
                v0 += bias[col0 + l16];
                v1 += bias[col0 + 16 + l16];
                v2 += bias[col0 + 32 + l16];
                v3 += bias[col0 + 48 + l16];
            }
            if (relu) {
                v0 = fmaxf(v0, 0.f); v1 = fmaxf(v1, 0.f);
                v2 = fmaxf(v2, 0.f); v3 = fmaxf(v3, 0.f);
            }
            C[base]      = v0;
            C[base + 16] = v1;
            C[base + 32] = v2;
            C[base + 48] = v3;
        }
    }
}

// ---------------- per-edge multi-head attention (softmax over head axis g) ----------------
// D = channels per head (32 for 256-dim convs, 8 for the 64-dim conv). H = 8 heads.
// D lanes cooperate on one edge; lane t holds element t of each head's q/k/v.
// Note: abias adds a per-h constant across g and cancels in the softmax over g.
template <int D>
__global__ void edge_attn(const float* __restrict__ Q, const float* __restrict__ Km,
                          const float* __restrict__ V,
                          const int* __restrict__ src, const int* __restrict__ dst,
                          float* __restrict__ OUT, long E, float scale)
{
    constexpr int H = 8;
    constexpr int C = H * D;
    constexpr int EPW = 32 / D;
    const int  lane = threadIdx.x & 31;
    const int  sub  = lane / D;
    const int  t    = lane % D;
    const long wave = ((long)blockIdx.x * blockDim.x + threadIdx.x) >> 5;
    const long e    = wave * EPW + sub;
    const bool valid = e < E;
    const int  s = valid ? src[e] : 0;
    const int  d = valid ? dst[e] : 0;

    float q[H], k[H], v[H];
#pragma unroll
    for (int h = 0; h < H; ++h) {
        q[h] = Q [(long)d * C + h * D + t];
        k[h] = Km[(long)s * C + h * D + t];
        v[h] = V [(long)s * C + h * D + t];
    }
    float sc[H][H];
#pragma unroll
    for (int h = 0; h < H; ++h)
#pragma unroll
        for (int g = 0; g < H; ++g) {
            float p = q[h] * k[g];
#pragma unroll
            for (int m = D >> 1; m >= 1; m >>= 1)
                p += __shfl_xor(p, m, 32);      // butterfly within the edge's D lanes
            sc[h][g] = p * scale;
        }
    if (!valid) return;
#pragma unroll
    for (int h = 0; h < H; ++h) {
        float mx = sc[h][0];
#pragma unroll
        for (int g = 1; g < H; ++g) mx = fmaxf(mx, sc[h][g]);
        float ssum = 0.f, o = 0.f;
#pragma unroll
        for (int g = 0; g < H; ++g) {
            float a = __expf(sc[h][g] - mx);
            ssum += a;
            o    += a * v[g];
        }
        atomicAdd(&OUT[(long)d * C + h * D + t], o / ssum);
    }
}

// ---------------- elementwise / scatter helpers ----------------
__global__ void fill_k(float* __restrict__ p, float v, long n)
{
    long i = (long)blockIdx.x * blockDim.x + threadIdx.x;
    const long st = (long)gridDim.x * blockDim.x;
    for (; i < n; i += st) p[i] = v;
}

__global__ void relu_k(float* __restrict__ p, long n)
{
    long i = (long)blockIdx.x * blockDim.x + threadIdx.x;
    const long st = (long)gridDim.x * blockDim.x;
    for (; i < n; i += st) p[i] = fmaxf(p[i], 0.f);
}

__global__ void deg_edge_k(const int* __restrict__ dst, float* __restrict__ deg, long E)
{
    long i = (long)blockIdx.x * blockDim.x + threadIdx.x;
    if (i < E) atomicAdd(&deg[dst[i]], 1.f);
}

__global__ void rsqrt_k(float* __restrict__ dis, const float* __restrict__ deg, long n)
{
    long i = (long)blockIdx.x * blockDim.x + threadIdx.x;
    if (i < n) dis[i] = rsqrtf(deg[i]);     // deg >= 1 always (self loop)
}

// OUT = H * dis^2 + bias   (self-loop term of GCN)
__global__ void gcn_init_k(const float* __restrict__ H, const float* __restrict__ dis,
                           const float* __restrict__ bias, float* __restrict__ OUT,
                           long N, int C)
{
    long i = (long)blockIdx.x * blockDim.x + threadIdx.x;
    const long st = (long)gridDim.x * blockDim.x;
    const long total = N * C;
    for (; i < total; i += st) {
        long n = i / C; int c = (int)(i - n * C);
        float d = dis[n];
        OUT[i] = H[i] * d * d + bias[c];
    }
}

// block-per-edge scatter: OUT[t] += H[s] * dis[s]*dis[t]
__global__ void gcn_scatter_k(const float* __restrict__ H, const float* __restrict__ dis,
                              const int* __restrict__ src, const int* __restrict__ dst,
                              float* __restrict__ OUT, int C)
{
    const int e = blockIdx.x;
    const int s = src[e], t = dst[e];
    const float w = dis[s] * dis[t];
    for (int c = threadIdx.x; c < C; c += blockDim.x)
        atomicAdd(&OUT[(long)t * C + c], H[(long)s * C + c] * w);
}

__global__ void pool_sum_k(const float* __restrict__ X, const int* __restrict__ seg,
                           float* __restrict__ sums, long N, int C)
{
    long i = (long)blockIdx.x * blockDim.x + threadIdx.x;
    const long st = (long)gridDim.x * blockDim.x;
    const long total = N * C;
    for (; i < total; i += st) {
        long n = i / C; int c = (int)(i - n * C);
        atomicAdd(&sums[(long)seg[n] * C + c], X[i]);
    }
}

__global__ void count_seg_k(const int* __restrict__ seg, float* __restrict__ cnt, long N)
{
    long i = (long)blockIdx.x * blockDim.x + threadIdx.x;
    if (i < N) atomicAdd(&cnt[seg[i]], 1.f);
}

// mean -> relu FC(64x64) -> GRUCell(h0=0); one block of 512 threads (b = tid/64, c = tid%64)
__global__ void sf_gru_k(const float* __restrict__ sfs, const float* __restrict__ cnt,
                         const float* __restrict__ sftw, const float* __restrict__ sftb,
                         const float* __restrict__ wih, const float* __restrict__ bih,
                         const float* __restrict__ bhh, float* __restrict__ sfg)
{
    __shared__ float s1[8 * 64];
    const int tid = threadIdx.x;
    const int b = tid >> 6, c = tid & 63;
    s1[tid] = sfs[tid] / fmaxf(cnt[b], 1.f);
    __syncthreads();
    float acc = sftb[c];
    for (int j = 0; j < 64; ++j) acc += s1[b * 64 + j] * sftw[j * 64 + c];
    acc = fmaxf(acc, 0.f);
    __syncthreads();
    s1[tid] = acc;
    __syncthreads();
    float gr = bih[c], gz = bih[64 + c], gn = bih[128 + c];
    for (int j = 0; j < 64; ++j) {
        const float x = s1[b * 64 + j];
        gr += x * wih[c * 64 + j];
        gz += x * wih[(64 + c) * 64 + j];
        gn += x * wih[(128 + c) * 64 + j];
    }
    const float r = sigmoidf_(gr + bhh[c]);
    const float z = sigmoidf_(gz + bhh[64 + c]);
    const float n = tanhf(gn + r * bhh[128 + c]);
    sfg[tid] = (1.f - z) * n;
}

// vf_wt[n, 0:64] = x64[n] + posenc(n);  vf_wt[n, 64:128] = sfg[batch[n]]
__global__ void build_vfwt_k(const float* __restrict__ x64, const float* __restrict__ sfg,
                             const int* __restrict__ batch, float* __restrict__ vf, long N)
{
    long i = (long)blockIdx.x * blockDim.x + threadIdx.x;
    const long st = (long)gridDim.x * blockDim.x;
    const long total = N * 128;
    for (; i < total; i += st) {
        long n = i >> 7; int c = (int)(i & 127);
        float v;
        if (c < 64) {
            const int id = c >> 1;
            const float div = __expf(-0.28782313662f * (float)id);  // exp(-ln(1e4)*2i/64)
            const float ang = (float)n * div;
            v = x64[n * 64 + c] + ((c & 1) ? __cosf(ang) : __sinf(ang));
        } else {
            v = sfg[(long)batch[n] * 64 + (c - 64)];
        }
        vf[i] = v;
    }
}

__global__ void hh_finalize_k(const float* __restrict__ gws, const float* __restrict__ cw,
                              const float* __restrict__ gds, const float* __restrict__ cd,
                              float* __restrict__ HH)
{
    const int idx = blockIdx.x * blockDim.x + threadIdx.x;   // 8 x 256
    if (idx >= 8 * 256) return;
    const int b = idx >> 8, c = idx & 255;
    HH[idx] = (c < 128) ? gws[b * 128 + c] / fmaxf(cw[b], 1.f)
                        : gds[b * 128 + (c - 128)] / fmaxf(cd[b], 1.f);
}

// tiny row GEMM for the B=8 head
__global__ void small_fc_k(const float* __restrict__ X, const float* __restrict__ W,
                           const float* __restrict__ b, float* __restrict__ Y,
                           int K, int M, int relu)
{
    const int m = blockIdx.x * blockDim.x + threadIdx.x;
    const int row = blockIdx.y;
    if (m >= M) return;
    float acc = b[m];
    for (int k = 0; k < K; ++k) acc += X[row * K + k] * W[k * M + m];
    if (relu) acc = fmaxf(acc, 0.f);
    Y[row * M + m] = acc;
}

// --------------------------------------------------------------------------------------
extern "C" void kernel_launch(void* const* d_in, const int* in_sizes, int n_in,
                              void* d_out, int out_size, void* d_ws, size_t ws_size,
                              hipStream_t stream)
{
    const float* x_wt   = (const float*)d_in[0];
    const float* x_diff = (const float*)d_in[1];
    const int*   ei_wt  = (const int*)d_in[2];
    const int*   ei_df  = (const int*)d_in[3];
    const int*   b_wt   = (const int*)d_in[4];
    const int*   b_df   = (const int*)d_in[5];

    const long N  = in_sizes[0] / 64;
    const long E  = in_sizes[2] / 2;
    const long Ed = in_sizes[3] / 2;
    const int *src_w = ei_wt, *dst_w = ei_wt + E;
    const int *src_d = ei_df, *dst_d = ei_df + Ed;

    const float* emb_w = (const float*)d_in[6];
    const float* emb_b = (const float*)d_in[7];
    const float* cw_lin_w = (const float*)d_in[8];
    const float* cw_lin_b = (const float*)d_in[9];
    const float* cw_q_w = (const float*)d_in[10];
    const float* cw_q_b = (const float*)d_in[11];
    const float* cw_k_w = (const float*)d_in[12];
    const float* cw_k_b = (const float*)d_in[13];
    const float* cw_v_w = (const float*)d_in[14];
    const float* cw_v_b = (const float*)d_in[15];
    /* d_in[16] = cw_bias : per-h additive constant, cancels in softmax over g */
    const float* c3_lin_w = (const float*)d_in[17];
    const float* c3_lin_b = (const float*)d_in[18];
    const float* c3_q_w = (const float*)d_in[19];
    const float* c3_q_b = (const float*)d_in[20];
    const float* c3_k_w = (const float*)d_in[21];
    const float* c3_k_b = (const float*)d_in[22];
    const float* c3_v_w = (const float*)d_in[23];
    const float* c3_v_b = (const float*)d_in[24];
    /* d_in[25] = c3_bias : cancels */
    const float* sft_w = (const float*)d_in[26];
    const float* sft_b = (const float*)d_in[27];
    const float* gru_wih = (const float*)d_in[28];
    /* d_in[29] = gru_whh : h0 == 0 -> contributes only bhh */
    const float* gru_bih = (const float*)d_in[30];
    const float* gru_bhh = (const float*)d_in[31];
    const float* gcn_w1 = (const float*)d_in[32];
    const float* gcn_b1 = (const float*)d_in[33];
    const float* gcn_w2 = (const float*)d_in[34];
    const float* gcn_b2 = (const float*)d_in[35];
    const float* gcn_w3 = (const float*)d_in[36];
    const float* gcn_b3 = (const float*)d_in[37];
    const float* mlp_w1 = (const float*)d_in[38];
    const float* mlp_b1 = (const float*)d_in[39];
    const float* mlp_w2 = (const float*)d_in[40];
    const float* mlp_b2 = (const float*)d_in[41];
    const float* mlp_w3 = (const float*)d_in[42];
    const float* mlp_b3 = (const float*)d_in[43];
    const float* pred_w = (const float*)d_in[44];
    const float* pred_b = (const float*)d_in[45];

    // ---- workspace carve-up (floats) ----
    float* W = (float*)d_ws;
    long off = 0;
    auto alloc = [&](long n) { float* p = W + off; off += n; return p; };
    float* B0  = alloc(N * 256);   // X / conv output accumulator
    float* B1  = alloc(N * 256);   // XL / GCN H
    float* B2  = alloc(N * 256);   // Q / GCN out
    float* B3  = alloc(N * 256);   // K
    float* B4  = alloc(N * 256);   // V
    float* VFW = alloc(N * 128);
    float* VFD = alloc(N * 128);
    float* DEG = alloc(N);
    float* DIS = alloc(N);
    float* SFS = alloc(8 * 64);
    float* CNT = alloc(8);
    float* SFG = alloc(8 * 64);
    float* GWS = alloc(8 * 128);
    float* GDS = alloc(8 * 128);
    float* CW  = alloc(8);
    float* CD  = alloc(8);
    float* HH  = alloc(8 * 256);
    float* M1  = alloc(8 * 128);
    float* M2  = alloc(8 * 128);
    float* M3  = alloc(8 * 128);
    (void)ws_size; (void)n_in; (void)out_size;

    auto gemm = [&](const float* A, const float* Bw, const float* bias, float* Cp,
                    long M, int K, int Nd, int relu) {
        const long tiles = ((M + 15) / 16) * (long)(Nd / 64);   // 16x64 tile per wave
        const int blocks = (int)((tiles + 7) / 8);              // 8 waves / 256-thread block
        gemm_wmma_f32<<<blocks, 256, 0, stream>>>(A, Bw, bias, Cp, (int)M, K, Nd, relu);
    };
    auto zero = [&](float* p, long n) {
        int blocks = (int)((n + 255) / 256); if (blocks > 16384) blocks = 16384;
        fill_k<<<blocks, 256, 0, stream>>>(p, 0.f, n);
    };
    auto relu_ip = [&](float* p, long n) {
        int blocks = (int)((n + 255) / 256); if (blocks > 16384) blocks = 16384;
        relu_k<<<blocks, 256, 0, stream>>>(p, n);
    };

    // =========== wild-type branch: embed + 2x CustomConv(256) + CustomConv(64) ===========
    gemm(x_wt, emb_w, emb_b, B0, N, 64, 256, 1);

    for (int i = 0; i < 2; ++i) {
        const long wOff = (long)i * 256 * 256, bOff = (long)i * 256;
        gemm(B0, cw_lin_w + wOff, cw_lin_b + bOff, B1, N, 256, 256, 0);
        gemm(B1, cw_q_w + wOff, cw_q_b + bOff, B2, N, 256, 256, 0);
        gemm(B1, cw_k_w + wOff, cw_k_b + bOff, B3, N, 256, 256, 0);
        gemm(B1, cw_v_w + wOff, cw_v_b + bOff, B4, N, 256, 256, 0);
        zero(B0, N * 256);
        edge_attn<32><<<(int)((E + 7) / 8), 256, 0, stream>>>(B2, B3, B4, src_w, dst_w,
                                                              B0, E, 0.1767766953f);
        relu_ip(B0, N * 256);
    }
    // conv3: 256 -> 64, H=8, d=8
    gemm(B0, c3_lin_w, c3_lin_b, B1, N, 256, 64, 0);
    gemm(B1, c3_q_w, c3_q_b, B2, N, 64, 64, 0);
    gemm(B1, c3_k_w, c3_k_b, B3, N, 64, 64, 0);
    gemm(B1, c3_v_w, c3_v_b, B4, N, 64, 64, 0);
    zero(B0, N * 64);
    edge_attn<8><<<(int)(((E + 3) / 4 + 7) / 8), 256, 0, stream>>>(B2, B3, B4, src_w, dst_w,
                                                                   B0, E, 0.35355339059f);
    relu_ip(B0, N * 64);

    // ---- segment mean -> FC -> GRU ----
    zero(SFS, 8 * 64 + 8);                                  // SFS and CNT are contiguous
    pool_sum_k<<<(int)((N * 64 + 255) / 256), 256, 0, stream>>>(B0, b_wt, SFS, N, 64);
    count_seg_k<<<(int)((N + 255) / 256), 256, 0, stream>>>(b_wt, CNT, N);
    sf_gru_k<<<1, 512, 0, stream>>>(SFS, CNT, sft_w, sft_b, gru_wih, gru_bih, gru_bhh, SFG);
    build_vfwt_k<<<(int)((N * 128 + 255) / 256), 256, 0, stream>>>(B0, SFG, b_wt, VFW, N);

    // =========== diff branch: 3x GCN ===========
    {
        int nb = (int)((N + 255) / 256);
        fill_k<<<nb, 256, 0, stream>>>(DEG, 1.f, N);        // self loop contributes 1
        deg_edge_k<<<(int)((Ed + 255) / 256), 256, 0, stream>>>(dst_d, DEG, Ed);
        rsqrt_k<<<nb, 256, 0, stream>>>(DIS, DEG, N);
    }
    // layer 1: 64 -> 256, relu
    gemm(x_diff, gcn_w1, nullptr, B1, N, 64, 256, 0);
    gcn_init_k<<<(int)((N * 256 + 255) / 256), 256, 0, stream>>>(B1, DIS, gcn_b1, B2, N, 256);
    gcn_scatter_k<<<(int)Ed, 256, 0, stream>>>(B1, DIS, src_d, dst_d, B2, 256);
    relu_ip(B2, N * 256);
    // layer 2: 256 -> 256, relu
    gemm(B2, gcn_w2, nullptr, B1, N, 256, 256, 0);
    gcn_init_k<<<(int)((N * 256 + 255) / 256), 256, 0, stream>>>(B1, DIS, gcn_b2, B2, N, 256);
    gcn_scatter_k<<<(int)Ed, 256, 0, stream>>>(B1, DIS, src_d, dst_d, B2, 256);
    relu_ip(B2, N * 256);
    // layer 3: 256 -> 128, no relu
    gemm(B2, gcn_w3, nullptr, B1, N, 256, 128, 0);
    gcn_init_k<<<(int)((N * 128 + 255) / 256), 256, 0, stream>>>(B1, DIS, gcn_b3, VFD, N, 128);
    gcn_scatter_k<<<(int)Ed, 128, 0, stream>>>(B1, DIS, src_d, dst_d, VFD, 128);

    // =========== pool + MLP head ===========
    zero(GWS, 8 * 128 * 2 + 16);                            // GWS, GDS, CW, CD contiguous
    pool_sum_k<<<(int)((N * 128 + 255) / 256), 256, 0, stream>>>(VFW, b_wt, GWS, N, 128);
    count_seg_k<<<(int)((N + 255) / 256), 256, 0, stream>>>(b_wt, CW, N);
    pool_sum_k<<<(int)((N * 128 + 255) / 256), 256, 0, stream>>>(VFD, b_df, GDS, N, 128);
    count_seg_k<<<(int)((N + 255) / 256), 256, 0, stream>>>(b_df, CD, N);
    hh_finalize_k<<<8, 256, 0, stream>>>(GWS, CW, GDS, CD, HH);

    small_fc_k<<<dim3(2, 8), 64, 0, stream>>>(HH, mlp_w1, mlp_b1, M1, 256, 128, 1);
    small_fc_k<<<dim3(2, 8), 64, 0, stream>>>(M1, mlp_w2, mlp_b2, M2, 128, 128, 1);
    small_fc_k<<<dim3(2, 8), 64, 0, stream>>>(M2, mlp_w3, mlp_b3, M3, 128, 128, 0);
    small_fc_k<<<dim3(1, 8), 64, 0, stream>>>(M3, pred_w, pred_b, (float*)d_out, 128, 1, 0);
}